// GCN_10771777978500
// MI455X (gfx1250) — compile-verified
//
#include <hip/hip_runtime.h>
#include <hip/hip_bf16.h>

// ---------------------------------------------------------------------------
// Hetero-GCN (DGL GraphConv, norm='both') for MI455X / gfx1250.
//   h_a: [NA,128] f32, h_b: [NB,64] f32, 4 layers.
//   Per layer: 3 row-scaled GEMMs (f32 WMMA 16x16x4), 3 edge scatter-adds
//   (f32 atomics, agg arrays L2-resident), fused scale+bias+relu+residual+LN.
// ---------------------------------------------------------------------------

#define NA_N 100000
#define NB_N 100000
#define E_N  1000000
#define DEPTH_N 4
#define LN_EPS 1e-3f

typedef __attribute__((ext_vector_type(2))) float v2f;
typedef __attribute__((ext_vector_type(8))) float v8f;

// ------------------------- degree kernels ---------------------------------
__global__ void gcn_count_deg(const int* __restrict__ src, const int* __restrict__ dst,
                              float* __restrict__ dout, float* __restrict__ din, int E) {
    int e = blockIdx.x * blockDim.x + threadIdx.x;
    if (e < E) {
        atomicAdd(&dout[src[e]], 1.0f);
        atomicAdd(&din[dst[e]], 1.0f);
    }
}

__global__ void gcn_rsqrt_deg(float* __restrict__ d, int n) {
    int i = blockIdx.x * blockDim.x + threadIdx.x;
    if (i < n) {
        float x = d[i];
        d[i] = rsqrtf(x > 1.0f ? x : 1.0f);
    }
}

// ------------------------- init kernels -----------------------------------
__global__ void gcn_init_ha(float* __restrict__ h, const float* __restrict__ feat,
                            const float* __restrict__ w_a, int n) {
    int i = blockIdx.x * blockDim.x + threadIdx.x;  // n = NA*128
    if (i < n) {
        int row = i >> 7, c = i & 127;
        h[i] = (c < 32) ? feat[row * 32 + c] : w_a[c - 32];
    }
}

__global__ void gcn_init_hb(float* __restrict__ h, const float* __restrict__ w_b, int n) {
    int i = blockIdx.x * blockDim.x + threadIdx.x;  // n = NB*64
    if (i < n) h[i] = w_b[i & 63];
}

// ------------------------- WMMA GEMM --------------------------------------
// C[N,64] = (A[N,K] * rowscale[N]) @ W[K,64]; one wave32 -> 16x64 strip.
// V_WMMA_F32_16X16X4_F32 layouts (ISA 7.12.2):
//   A 16x4 : lane m=L&15; VGPR0/1 = K (L<16 ? k,k+1 : k+2,k+3)
//   B 4x16 : n=L&15;      VGPR0 = K (L<16 ? k : k+2), VGPR1 = K+1 of same half
//   C/D    : VGPR r -> row r (+8 for L>=16), col = L&15
// W is staged in LDS interleaved as K-pairs so each B fragment is one aligned
// ds_load_b64: Wl[(k>>1)*PP + n*2 + (k&1)]. PP = 160 dwords (128 data + 32
// pad) puts the two half-waves (k-pair rows kp and kp+1) on disjoint bank
// halves (banks 0-31 vs 32-63) within a single b64 load.
template <int K>
__launch_bounds__(256)
__global__ void gcn_gemm_wmma(const float* __restrict__ A, const float* __restrict__ rowscale,
                              const float* __restrict__ W, float* __restrict__ C, int N) {
    constexpr int PP = 160;  // dword pitch per k-pair row
    __shared__ float Wl[(K / 2) * PP];
    const int tid = threadIdx.x;
    for (int i = tid; i < K * 64; i += 256) {
        int r = i >> 6, c = i & 63;
        Wl[(r >> 1) * PP + (c << 1) + (r & 1)] = W[i];
    }
    __syncthreads();

    const int lane    = tid & 31;
    const int wave    = tid >> 5;
    const int rowBase = (blockIdx.x * 8 + wave) << 4;
    if (rowBase >= N) return;  // whole-wave uniform: EXEC stays all-ones for WMMA

    const int m     = lane & 15;
    const int half  = lane >> 4;        // 0 or 1
    const int koff  = half << 1;        // 0 or 2
    const float s   = rowscale[rowBase + m];
    const float* Arow = A + (size_t)(rowBase + m) * K;

    v8f acc0 = {}, acc1 = {}, acc2 = {}, acc3 = {};
    for (int k = 0; k < K; k += 4) {
        const int ko = k + koff;
        v2f a;
        a.x = Arow[ko] * s;
        a.y = Arow[ko + 1] * s;
        const v2f* wp = (const v2f*)&Wl[(ko >> 1) * PP + (m << 1)];
        v2f b0 = wp[0];
        v2f b1 = wp[16];
        v2f b2 = wp[32];
        v2f b3 = wp[48];
        acc0 = __builtin_amdgcn_wmma_f32_16x16x4_f32(false, a, false, b0, (short)0, acc0, false, false);
        acc1 = __builtin_amdgcn_wmma_f32_16x16x4_f32(false, a, false, b1, (short)0, acc1, false, false);
        acc2 = __builtin_amdgcn_wmma_f32_16x16x4_f32(false, a, false, b2, (short)0, acc2, false, false);
        acc3 = __builtin_amdgcn_wmma_f32_16x16x4_f32(false, a, false, b3, (short)0, acc3, false, false);
    }

    const int rofs = half << 3;  // 0 or 8
#pragma unroll
    for (int r = 0; r < 8; ++r) {
        float* Crow = C + (size_t)(rowBase + rofs + r) * 64 + m;
        Crow[0]  = acc0[r];
        Crow[16] = acc1[r];
        Crow[32] = acc2[r];
        Crow[48] = acc3[r];
    }
}

// ------------------------- edge scatter-add -------------------------------
// 1 thread per (edge, col): coalesced row gather, f32 atomic scatter (L2-resident agg).
__global__ void gcn_scatter64(const float* __restrict__ proj, const int* __restrict__ src,
                              const int* __restrict__ dst, float* __restrict__ agg,
                              int pitch, int E) {
    int tid = blockIdx.x * blockDim.x + threadIdx.x;  // E*64 threads
    int e = tid >> 6;
    int c = tid & 63;
    if (e < E) {
        int s = src[e], d = dst[e];
        atomicAdd(&agg[(size_t)d * pitch + c], proj[(size_t)s * 64 + c]);
    }
}

// ------------------------- finalize + LayerNorm ---------------------------
// h_a = LN( [relu(agg0*rs_aa+b_aa) | relu(agg1*rs_ba+b_ba)] + h_a , beta ); wave32/row.
__global__ void gcn_finalize_a(float* __restrict__ h, const float* __restrict__ agg,
                               const float* __restrict__ rs_in_aa, const float* __restrict__ rs_in_ba,
                               const float* __restrict__ bias_aa, const float* __restrict__ bias_ba,
                               const float* __restrict__ beta, int n) {
    int lane = threadIdx.x & 31, wave = threadIdx.x >> 5;
    int row = blockIdx.x * 8 + wave;
    if (row >= n) return;
    const float* arow = agg + (size_t)row * 128;
    float* hrow = h + (size_t)row * 128;
    int c0 = lane * 4;
    float rsv = (c0 < 64) ? rs_in_aa[row] : rs_in_ba[row];  // lane-uniform halves
    float v[4];
    float sum = 0.0f;
#pragma unroll
    for (int j = 0; j < 4; ++j) {
        int c = c0 + j;
        float b = (c < 64) ? bias_aa[c] : bias_ba[c - 64];
        float t = fmaf(arow[c], rsv, b);
        t = fmaxf(t, 0.0f);
        t += hrow[c];
        v[j] = t;
        sum += t;
    }
#pragma unroll
    for (int off = 16; off; off >>= 1) sum += __shfl_xor(sum, off, 32);
    float mu = sum * (1.0f / 128.0f);
    float var = 0.0f;
#pragma unroll
    for (int j = 0; j < 4; ++j) { float d = v[j] - mu; var += d * d; }
#pragma unroll
    for (int off = 16; off; off >>= 1) var += __shfl_xor(var, off, 32);
    float inv = rsqrtf(var * (1.0f / 128.0f) + LN_EPS);
#pragma unroll
    for (int j = 0; j < 4; ++j) {
        int c = c0 + j;
        hrow[c] = (v[j] - mu) * inv + beta[c];
    }
}

__global__ void gcn_finalize_b(float* __restrict__ h, const float* __restrict__ agg,
                               const float* __restrict__ rs_in, const float* __restrict__ bias,
                               const float* __restrict__ beta, int n) {
    int lane = threadIdx.x & 31, wave = threadIdx.x >> 5;
    int row = blockIdx.x * 8 + wave;
    if (row >= n) return;
    const float* arow = agg + (size_t)row * 64;
    float* hrow = h + (size_t)row * 64;
    float rsv = rs_in[row];
    int c0 = lane * 2;
    float v[2];
    float sum = 0.0f;
#pragma unroll
    for (int j = 0; j < 2; ++j) {
        int c = c0 + j;
        float t = fmaf(arow[c], rsv, bias[c]);
        t = fmaxf(t, 0.0f);
        t += hrow[c];
        v[j] = t;
        sum += t;
    }
#pragma unroll
    for (int off = 16; off; off >>= 1) sum += __shfl_xor(sum, off, 32);
    float mu = sum * (1.0f / 64.0f);
    float var = 0.0f;
#pragma unroll
    for (int j = 0; j < 2; ++j) { float d = v[j] - mu; var += d * d; }
#pragma unroll
    for (int off = 16; off; off >>= 1) var += __shfl_xor(var, off, 32);
    float inv = rsqrtf(var * (1.0f / 64.0f) + LN_EPS);
#pragma unroll
    for (int j = 0; j < 2; ++j) {
        int c = c0 + j;
        hrow[c] = (v[j] - mu) * inv + beta[c];
    }
}

// ---------------------------------------------------------------------------
extern "C" void kernel_launch(void* const* d_in, const int* in_sizes, int n_in,
                              void* d_out, int out_size, void* d_ws, size_t ws_size,
                              hipStream_t stream) {
    (void)in_sizes; (void)n_in; (void)out_size; (void)ws_size;
    const int NA = NA_N, NB = NB_N, E = E_N;

    // ---- inputs (setup_inputs() dict insertion order) ----
    const float* feat_a = (const float*)d_in[0];
    const float* w_a    = (const float*)d_in[1];
    const float* w_b    = (const float*)d_in[2];
    const float *W_aa[DEPTH_N], *b_aa[DEPTH_N], *W_ba[DEPTH_N], *b_ba[DEPTH_N];
    const float *W_ab[DEPTH_N], *b_ab[DEPTH_N], *beta_a[DEPTH_N], *beta_b[DEPTH_N];
    for (int l = 0; l < DEPTH_N; ++l) {
        int base = 3 + l * 8;
        W_aa[l]   = (const float*)d_in[base + 0];
        b_aa[l]   = (const float*)d_in[base + 1];
        W_ba[l]   = (const float*)d_in[base + 2];
        b_ba[l]   = (const float*)d_in[base + 3];
        W_ab[l]   = (const float*)d_in[base + 4];
        b_ab[l]   = (const float*)d_in[base + 5];
        beta_a[l] = (const float*)d_in[base + 6];
        beta_b[l] = (const float*)d_in[base + 7];
    }
    const int* src_aa = (const int*)d_in[35];
    const int* dst_aa = (const int*)d_in[36];
    const int* src_ba = (const int*)d_in[37];
    const int* dst_ba = (const int*)d_in[38];
    const int* src_ab = (const int*)d_in[39];
    const int* dst_ab = (const int*)d_in[40];

    // ---- outputs double as live h buffers across layers ----
    float* h_a = (float*)d_out;                      // [NA,128]
    float* h_b = (float*)d_out + (size_t)NA * 128;   // [NB,64]

    // ---- workspace layout ----
    float* ws       = (float*)d_ws;
    float* agg_a    = ws;                                   // NA*128
    float* agg_b    = agg_a + (size_t)NA * 128;             // NB*64
    float* proj_aa  = agg_b + (size_t)NB * 64;              // NA*64
    float* proj_ba  = proj_aa + (size_t)NA * 64;            // NB*64
    float* proj_ab  = proj_ba + (size_t)NB * 64;            // NA*64
    float* rs_out_aa = proj_ab + (size_t)NA * 64;           // 6 x 100000, contiguous
    float* rs_in_aa  = rs_out_aa + NA;
    float* rs_out_ba = rs_in_aa + NA;
    float* rs_in_ba  = rs_out_ba + NB;
    float* rs_out_ab = rs_in_ba + NA;
    float* rs_in_ab  = rs_out_ab + NA;

    // ---- degrees -> rsqrt(max(deg,1)) (graph is fixed; cheap, done per call) ----
    hipMemsetAsync(rs_out_aa, 0, (size_t)6 * 100000 * sizeof(float), stream);
    int eb = (E + 255) / 256;
    gcn_count_deg<<<eb, 256, 0, stream>>>(src_aa, dst_aa, rs_out_aa, rs_in_aa, E);
    gcn_count_deg<<<eb, 256, 0, stream>>>(src_ba, dst_ba, rs_out_ba, rs_in_ba, E);
    gcn_count_deg<<<eb, 256, 0, stream>>>(src_ab, dst_ab, rs_out_ab, rs_in_ab, E);
    gcn_rsqrt_deg<<<(6 * 100000 + 255) / 256, 256, 0, stream>>>(rs_out_aa, 6 * 100000);

    // ---- initial embeddings ----
    gcn_init_ha<<<(NA * 128) / 256, 256, 0, stream>>>(h_a, feat_a, w_a, NA * 128);
    gcn_init_hb<<<(NB * 64) / 256, 256, 0, stream>>>(h_b, w_b, NB * 64);

    const int gemm_blocks = (NA / 16 + 7) / 8;   // 782 (NA == NB)
    const int scat_blocks = (E * 64) / 256;      // 250000
    const int fin_blocks  = NA / 8;              // 12500 (NA == NB)

    for (int l = 0; l < DEPTH_N; ++l) {
        // zero agg_a|agg_b (contiguous)
        hipMemsetAsync(agg_a, 0, ((size_t)NA * 128 + (size_t)NB * 64) * sizeof(float), stream);

        // projections (deg_out^{-1/2} fused into A load)
        gcn_gemm_wmma<128><<<gemm_blocks, 256, 0, stream>>>(h_a, rs_out_aa, W_aa[l], proj_aa, NA);
        gcn_gemm_wmma<128><<<gemm_blocks, 256, 0, stream>>>(h_a, rs_out_ab, W_ab[l], proj_ab, NA);
        gcn_gemm_wmma<64><<<gemm_blocks, 256, 0, stream>>>(h_b, rs_out_ba, W_ba[l], proj_ba, NB);

        // edge scatter: o_aa -> agg_a[:, :64], o_ba -> agg_a[:, 64:], o_ab -> agg_b
        gcn_scatter64<<<scat_blocks, 256, 0, stream>>>(proj_aa, src_aa, dst_aa, agg_a, 128, E);
        gcn_scatter64<<<scat_blocks, 256, 0, stream>>>(proj_ba, src_ba, dst_ba, agg_a + 64, 128, E);
        gcn_scatter64<<<scat_blocks, 256, 0, stream>>>(proj_ab, src_ab, dst_ab, agg_b, 64, E);

        // deg_in^{-1/2} + bias + relu + residual + LayerNorm(beta)
        gcn_finalize_a<<<fin_blocks, 256, 0, stream>>>(h_a, agg_a, rs_in_aa, rs_in_ba,
                                                       b_aa[l], b_ba[l], beta_a[l], NA);
        gcn_finalize_b<<<fin_blocks, 256, 0, stream>>>(h_b, agg_b, rs_in_ab,
                                                       b_ab[l], beta_b[l], NB);
    }
}